// ResidualBlock_59768764891689
// MI455X (gfx1250) — compile-verified
//
#include <hip/hip_runtime.h>
#include <stdint.h>

typedef __attribute__((ext_vector_type(2))) float v2f;
typedef __attribute__((ext_vector_type(8))) float v8f;

#define CCH 96
#define KOFF 27
#define TILE_ROWS 32
#define TILES_PER_BLOCK 10
#define CONV_THREADS 384   // 12 wave32 waves
#define BN_EPS 1e-5f

// ---------------------------------------------------------------------------
// Sparse conv for one kernel offset k (blockIdx.y). Each block stages W[k]
// (96x96, 36KB) in LDS once, then loops over TILES_PER_BLOCK row-tiles of 32
// gathered+masked rows with a double-buffered LDS stage: gather for tile t+1
// overlaps the 24-deep fp32 WMMA chain for tile t. Scatter-add via global
// f32 atomics (out_idx is random; collisions across blocks are expected).
// ---------------------------------------------------------------------------
__global__ __launch_bounds__(CONV_THREADS)
void sparse_conv_wmma(const float* __restrict__ x,
                      const float* __restrict__ W,
                      const int* __restrict__ in_idx,
                      const int* __restrict__ out_idx,
                      const uint8_t* __restrict__ mask,
                      float* __restrict__ acc,
                      int n) {
  __shared__ float sW[CCH * CCH];               // 36 KB: W[k]
  __shared__ float sA[2][TILE_ROWS * CCH];      // 2 x 12 KB: gathered rows

  const int k = blockIdx.y;
  const int tile0 = blockIdx.x * TILES_PER_BLOCK;
  const float* Wk = W + (size_t)k * CCH * CCH;
  const size_t kn = (size_t)k * (size_t)n;

  // Stage weights once per block (9216 floats / 384 threads = 24 each).
  for (int i = threadIdx.x; i < CCH * CCH; i += CONV_THREADS) sW[i] = Wk[i];

  // Gather+mask one 32-row tile into an LDS buffer (float4 vectorized).
  auto stage_tile = [&](int tile, int buf) {
    const int rowbase = tile * TILE_ROWS;
    for (int i = threadIdx.x; i < TILE_ROWS * (CCH / 4); i += CONV_THREADS) {
      int r  = i / (CCH / 4);
      int cq = i % (CCH / 4);
      int g  = rowbase + r;
      int gs = (g < n) ? g : (n - 1);
      int src = in_idx[kn + gs];
      float m = (g < n && mask[kn + gs]) ? 1.0f : 0.0f;
      float4 v = ((const float4*)(x + (size_t)src * CCH))[cq];
      v.x *= m; v.y *= m; v.z *= m; v.w *= m;
      ((float4*)(sA[buf] + r * CCH))[cq] = v;
    }
  };

  // Per-wave tile assignment: 2 row-subtiles x 6 col-tiles.
  const int wave   = threadIdx.x >> 5;
  const int lane   = threadIdx.x & 31;
  const int rowsub = wave / 6;    // 0..1
  const int ct     = wave % 6;    // 0..5
  const int m16    = lane & 15;
  const int hi     = lane >> 4;
  // B (4x16): VGPR0 -> K = 2*hi, VGPR1 -> K = 2*hi+1, N = m16.
  const float* Bcol = sW + (2 * hi) * CCH + ct * 16 + m16;
  const int colbase = ct * 16 + m16;

  stage_tile(tile0, 0);
  __syncthreads();

  for (int t = 0; t < TILES_PER_BLOCK; ++t) {
    const int tile = tile0 + t;
    const int rowbase = tile * TILE_ROWS;
    if (t + 1 < TILES_PER_BLOCK) stage_tile(tile0 + t + 1, (t + 1) & 1);

    if (rowbase < n) {  // block-uniform; EXEC stays all-1s for WMMA
      // fp32 16x4 A layout: lanes 0-15 hold K=0,1 ; lanes 16-31 hold K=2,3.
      const float* Arow = sA[t & 1] + (rowsub * 16 + m16) * CCH + 2 * hi;

      v8f c = {};
#pragma unroll
      for (int ks = 0; ks < CCH / 4; ++ks) {
        v2f a, b;
        a.x = Arow[ks * 4 + 0];
        a.y = Arow[ks * 4 + 1];
        b.x = Bcol[(ks * 4 + 0) * CCH];
        b.y = Bcol[(ks * 4 + 1) * CCH];
        c = __builtin_amdgcn_wmma_f32_16x16x4_f32(
            /*neg_a=*/false, a, /*neg_b=*/false, b,
            /*c_mod=*/(short)0, c, /*reuse_a=*/false, /*reuse_b=*/false);
      }

      // C/D layout: VGPR r -> (lanes 0-15: M=r) (lanes 16-31: M=r+8), N=m16.
#pragma unroll
      for (int r = 0; r < 8; ++r) {
        int M = r + hi * 8;
        int g = rowbase + rowsub * 16 + M;
        if (g < n) {
          int dst = out_idx[kn + g];
          atomicAdd(acc + (size_t)dst * CCH + colbase, c[r]);
        }
      }
    }
    __syncthreads();  // staging(t+1) done for all; compute(t) done before reuse
  }
}

// ---------------------------------------------------------------------------
// Helpers
// ---------------------------------------------------------------------------
__global__ void zero_f32(float* __restrict__ p, size_t total) {
  size_t i = (size_t)blockIdx.x * blockDim.x + threadIdx.x;
  size_t stride = (size_t)gridDim.x * blockDim.x;
  for (; i < total; i += stride) p[i] = 0.0f;
}

// 96 threads/block, thread = channel; coalesced column reads + partial sums.
__global__ void bn_stats(const float* __restrict__ acc,
                         float* __restrict__ sum, float* __restrict__ sumsq,
                         int n, int rows_per_block) {
  int c  = threadIdx.x;
  int r0 = blockIdx.x * rows_per_block;
  int r1 = r0 + rows_per_block; if (r1 > n) r1 = n;
  float s = 0.0f, s2 = 0.0f;
  for (int r = r0; r < r1; ++r) {
    float v = acc[(size_t)r * CCH + c];
    s += v; s2 += v * v;
  }
  atomicAdd(&sum[c], s);
  atomicAdd(&sumsq[c], s2);
}

__global__ void bn_finalize(const float* __restrict__ sum,
                            const float* __restrict__ sumsq,
                            const float* __restrict__ gamma,
                            const float* __restrict__ beta,
                            float* __restrict__ scale,
                            float* __restrict__ bias, int n) {
  int c = threadIdx.x;
  if (c < CCH) {
    float inv_n = 1.0f / (float)n;
    float mu  = sum[c] * inv_n;
    float var = sumsq[c] * inv_n - mu * mu;   // biased variance, as reference
    float sc  = gamma[c] * rsqrtf(var + BN_EPS);
    scale[c] = sc;
    bias[c]  = beta[c] - mu * sc;
  }
}

// total4 = N*C/4 float4 elements; channel base = (4*i) % 96 (96 % 4 == 0).
__global__ void bn_relu_inplace(float4* __restrict__ buf,
                                const float* __restrict__ scale,
                                const float* __restrict__ bias, size_t total4) {
  size_t i = (size_t)blockIdx.x * blockDim.x + threadIdx.x;
  if (i < total4) {
    int cb = (int)((i * 4) % CCH);
    float4 v = buf[i];
    v.x = fmaxf(v.x * scale[cb + 0] + bias[cb + 0], 0.0f);
    v.y = fmaxf(v.y * scale[cb + 1] + bias[cb + 1], 0.0f);
    v.z = fmaxf(v.z * scale[cb + 2] + bias[cb + 2], 0.0f);
    v.w = fmaxf(v.w * scale[cb + 3] + bias[cb + 3], 0.0f);
    buf[i] = v;
  }
}

__global__ void bn_residual_relu(const float4* __restrict__ acc,
                                 const float4* __restrict__ x,
                                 const float* __restrict__ scale,
                                 const float* __restrict__ bias,
                                 float4* __restrict__ out, size_t total4) {
  size_t i = (size_t)blockIdx.x * blockDim.x + threadIdx.x;
  if (i < total4) {
    int cb = (int)((i * 4) % CCH);
    float4 a = acc[i], r = x[i], v;
    v.x = fmaxf(a.x * scale[cb + 0] + bias[cb + 0] + r.x, 0.0f);
    v.y = fmaxf(a.y * scale[cb + 1] + bias[cb + 1] + r.y, 0.0f);
    v.z = fmaxf(a.z * scale[cb + 2] + bias[cb + 2] + r.z, 0.0f);
    v.w = fmaxf(a.w * scale[cb + 3] + bias[cb + 3] + r.w, 0.0f);
    out[i] = v;
  }
}

// ---------------------------------------------------------------------------
// Launcher
// ---------------------------------------------------------------------------
extern "C" void kernel_launch(void* const* d_in, const int* in_sizes, int n_in,
                              void* d_out, int out_size, void* d_ws, size_t ws_size,
                              hipStream_t stream) {
  const float*   x      = (const float*)d_in[0];
  const float*   W1     = (const float*)d_in[1];
  const float*   gamma1 = (const float*)d_in[2];
  const float*   beta1  = (const float*)d_in[3];
  const float*   W2     = (const float*)d_in[4];
  const float*   gamma2 = (const float*)d_in[5];
  const float*   beta2  = (const float*)d_in[6];
  const int*     in_idx = (const int*)d_in[7];
  const int*     out_idx= (const int*)d_in[8];
  const uint8_t* mask   = (const uint8_t*)d_in[9];  // jnp.bool_ -> 1 byte/elt

  const int n = in_sizes[0] / CCH;                  // 200000 active voxels
  const size_t nc = (size_t)n * CCH;

  float* buf0 = (float*)d_ws;        // conv1 acc, then hidden activations
  float* buf1 = buf0 + nc;           // conv2 acc
  float* st   = buf1 + nc;           // sum1,sq1,sc1,bi1,sum2,sq2,sc2,bi2 (8*96)

  // Zero all accumulators + stats (ws is poisoned; atomics need zeros).
  zero_f32<<<2048, 256, 0, stream>>>(buf0, 2 * nc + 8 * CCH);

  const int tiles = (n + TILE_ROWS - 1) / TILE_ROWS;
  dim3 cgrid((unsigned)((tiles + TILES_PER_BLOCK - 1) / TILES_PER_BLOCK), KOFF);
  const int RPB = 512;
  const int sblocks = (n + RPB - 1) / RPB;
  const size_t nc4 = nc / 4;
  const int eblocks = (int)((nc4 + 255) / 256);

  // conv1 -> BN1 -> ReLU (in place in buf0)
  sparse_conv_wmma<<<cgrid, CONV_THREADS, 0, stream>>>(x, W1, in_idx, out_idx, mask, buf0, n);
  bn_stats<<<sblocks, CCH, 0, stream>>>(buf0, st, st + 96, n, RPB);
  bn_finalize<<<1, CCH, 0, stream>>>(st, st + 96, gamma1, beta1, st + 192, st + 288, n);
  bn_relu_inplace<<<eblocks, 256, 0, stream>>>((float4*)buf0, st + 192, st + 288, nc4);

  // conv2 -> BN2 -> +residual -> ReLU
  sparse_conv_wmma<<<cgrid, CONV_THREADS, 0, stream>>>(buf0, W2, in_idx, out_idx, mask, buf1, n);
  bn_stats<<<sblocks, CCH, 0, stream>>>(buf1, st + 384, st + 480, n, RPB);
  bn_finalize<<<1, CCH, 0, stream>>>(st + 384, st + 480, gamma2, beta2, st + 576, st + 672, n);
  bn_residual_relu<<<eblocks, 256, 0, stream>>>((float4*)buf1, (const float4*)x,
                                                st + 576, st + 672, (float4*)d_out, nc4);
}